// Pull_37091337568590
// MI455X (gfx1250) — compile-verified
//
#include <hip/hip_runtime.h>

// Pull2D with wrap (circulant) boundary, bilinear interpolation.
// x:   [B, C, H, W] float32   (C=16, H=W=512)
// phi: [B, 2, H, W] float32   (plane 0 = dy, plane 1 = dx, in voxels)
// out: [B, C, H, W] float32
//
// One thread per (b,h,w); loops all C channels so the index/weight math and
// the phi reads are amortized across channels. Powers-of-two H/W turn the
// wrap into '& 511'. Output uses non-temporal stores so the streaming output
// doesn't evict the x planes (which have ~4x bilinear reuse) from L2.

#define PW_H    512
#define PW_W    512
#define PW_C    16
#define PW_HW   (PW_H * PW_W)        // 2^18
#define PW_MASK 511

__global__ __launch_bounds__(256) void pull2d_wrap_kernel(
    const float* __restrict__ x,
    const float* __restrict__ phi,
    float* __restrict__ out,
    int total)                        // B * H * W
{
    int idx = blockIdx.x * blockDim.x + threadIdx.x;
    if (idx >= total) return;

    // Decompose idx -> (b, h, w); all powers of two -> shifts/masks only.
    int b   = idx >> 18;              // / (H*W)
    int pix = idx & (PW_HW - 1);      // h*W + w
    int h   = pix >> 9;               // / W
    int w   = pix & PW_MASK;          // % W

    // phi reads: exactly one thread touches each element -> non-temporal.
    const float* phib = phi + (size_t)b * 2 * PW_HW;
    float dy = __builtin_nontemporal_load(phib + pix);            // phi[b,0,h,w]
    float dx = __builtin_nontemporal_load(phib + PW_HW + pix);    // phi[b,1,h,w]

    float cy = dy + (float)h;
    float cx = dx + (float)w;
    float y0f = floorf(cy);
    float x0f = floorf(cx);
    float wy = cy - y0f;
    float wx = cx - x0f;

    // Wrap: & (2^k - 1) == mod for negative ints in two's complement.
    int y0 = ((int)y0f) & PW_MASK;
    int x0 = ((int)x0f) & PW_MASK;
    int y1 = (y0 + 1) & PW_MASK;
    int x1 = (x0 + 1) & PW_MASK;

    // In-plane element offsets for the 4 bilinear taps.
    int o00 = (y0 << 9) | x0;
    int o01 = (y0 << 9) | x1;
    int o10 = (y1 << 9) | x0;
    int o11 = (y1 << 9) | x1;

    float wx1 = 1.0f - wx;
    float wy1 = 1.0f - wy;

    const float* xb = x   + (size_t)b * PW_C * PW_HW;
    float*       ob = out + (size_t)b * PW_C * PW_HW + pix;

    // Full unroll: compiler clusters the 64 gathers into load clauses,
    // maximizing memory-level parallelism before the first s_wait_loadcnt.
#pragma unroll
    for (int c = 0; c < PW_C; ++c) {
        const float* xc = xb + c * PW_HW;
        float v00 = xc[o00];
        float v01 = xc[o01];
        float v10 = xc[o10];
        float v11 = xc[o11];
        // Match reference lerp ordering: top/bot then vertical.
        float top = v00 * wx1 + v01 * wx;
        float bot = v10 * wx1 + v11 * wx;
        float r   = top * wy1 + bot * wy;
        // Streaming output: non-temporal so we don't evict x from L2.
        __builtin_nontemporal_store(r, ob + c * PW_HW);
    }
}

extern "C" void kernel_launch(void* const* d_in, const int* in_sizes, int n_in,
                              void* d_out, int out_size, void* d_ws, size_t ws_size,
                              hipStream_t stream) {
    const float* x   = (const float*)d_in[0];   // [B,16,512,512]
    const float* phi = (const float*)d_in[1];   // [B,2,512,512]
    float*       out = (float*)d_out;           // [B,16,512,512]

    int B = in_sizes[0] / (PW_C * PW_HW);       // robust to batch size
    int total = B * PW_HW;                      // threads: one per (b,h,w)

    const int block = 256;                      // 8 wave32 waves per block
    int grid = (total + block - 1) / block;

    pull2d_wrap_kernel<<<grid, block, 0, stream>>>(x, phi, out, total);
}